// customLoss_16776142258912
// MI455X (gfx1250) — compile-verified
//
#include <hip/hip_runtime.h>
#include <math.h>

#define NROWS 4096
#define NCOLS 50257
#define BLK   512   // 16 wave32 per row-block

typedef __attribute__((ext_vector_type(2))) float v2f;
typedef __attribute__((ext_vector_type(8))) float v8f;

__device__ __forceinline__ float waveSum(float v) {
#pragma unroll
  for (int off = 16; off >= 1; off >>= 1) v += __shfl_xor(v, off, 32);
  return v;
}
__device__ __forceinline__ float waveMax(float v) {
#pragma unroll
  for (int off = 16; off >= 1; off >>= 1) v = fmaxf(v, __shfl_xor(v, off, 32));
  return v;
}

// One workgroup per row. Pass A: row max + masked sum (label entry negated).
// Pass B: sum exp(x - m) (row re-read hits L2: 196KB/row << 192MB).
__global__ __launch_bounds__(BLK) void row_pass_kernel(
    const float* __restrict__ out, const long long* __restrict__ labels,
    float* __restrict__ ws_masked, float* __restrict__ ws_nll) {
  const int row  = blockIdx.x;
  const int tid  = threadIdx.x;
  const size_t base = (size_t)row * NCOLS;
  const float* x = out + base;
  const int label = (int)labels[row];

  // row base is only 4B-aligned (NCOLS odd): scalar prologue, then float4 body.
  const int pre  = (int)((4u - ((unsigned)base & 3u)) & 3u);
  const int nvec = (NCOLS - pre) >> 2;
  const int tail = pre + (nvec << 2);

  float lmax = -INFINITY, lmask = 0.0f;

  if (tid < pre) {
    float v = x[tid];
    lmax = fmaxf(lmax, v);
    float t = (tid == label) ? -v : v;
    if (t > -1.0f) lmask += t + 2.0f;
  }
  const float4* xv = reinterpret_cast<const float4*>(x + pre);
  for (int k = tid; k < nvec; k += BLK) {
    float4 v = xv[k];
    const int col = pre + (k << 2);
    float e0 = v.x, e1 = v.y, e2 = v.z, e3 = v.w;
    lmax = fmaxf(fmaxf(fmaxf(lmax, e0), fmaxf(e1, e2)), e3);
    float t0 = (col + 0 == label) ? -e0 : e0;
    float t1 = (col + 1 == label) ? -e1 : e1;
    float t2 = (col + 2 == label) ? -e2 : e2;
    float t3 = (col + 3 == label) ? -e3 : e3;
    if (t0 > -1.0f) lmask += t0 + 2.0f;
    if (t1 > -1.0f) lmask += t1 + 2.0f;
    if (t2 > -1.0f) lmask += t2 + 2.0f;
    if (t3 > -1.0f) lmask += t3 + 2.0f;
  }
  for (int col = tail + tid; col < NCOLS; col += BLK) {
    float v = x[col];
    lmax = fmaxf(lmax, v);
    float t = (col == label) ? -v : v;
    if (t > -1.0f) lmask += t + 2.0f;
  }

  __shared__ float s_tmp[16];
  __shared__ float s_bcast;
  const int wave = tid >> 5, lane = tid & 31;

  // block max
  float wm = waveMax(lmax);
  if (lane == 0) s_tmp[wave] = wm;
  __syncthreads();
  if (wave == 0) {
    float m = (lane < 16) ? s_tmp[lane] : -INFINITY;
    m = waveMax(m);
    if (lane == 0) s_bcast = m;
  }
  __syncthreads();
  const float m = s_bcast;
  __syncthreads();                       // wave0 done with s_tmp before reuse

  // block masked sum
  float wk = waveSum(lmask);
  if (lane == 0) s_tmp[wave] = wk;
  __syncthreads();
  float masked_total = 0.0f;
  if (wave == 0) {
    float s = (lane < 16) ? s_tmp[lane] : 0.0f;
    masked_total = waveSum(s);
  }
  __syncthreads();                       // wave0 done with s_tmp before reuse

  // pass B: one exp per element
  float lsum = 0.0f;
  if (tid < pre) lsum += __expf(x[tid] - m);
  for (int k = tid; k < nvec; k += BLK) {
    float4 v = xv[k];
    lsum += __expf(v.x - m) + __expf(v.y - m) + __expf(v.z - m) + __expf(v.w - m);
  }
  for (int col = tail + tid; col < NCOLS; col += BLK) lsum += __expf(x[col] - m);

  float ws = waveSum(lsum);
  if (lane == 0) s_tmp[wave] = ws;
  __syncthreads();
  if (wave == 0) {
    float s = (lane < 16) ? s_tmp[lane] : 0.0f;
    s = waveSum(s);
    if (lane == 0) {
      float lse = m + __logf(s);         // double log_softmax is idempotent
      ws_masked[row] = masked_total;
      ws_nll[row]    = lse - x[label];
    }
  }
}

// Final reduction of 4096 partials with V_WMMA_F32_16X16X4_F32 (ones B-matrix):
// D[m][n] = sum_k A[m][k], accumulated into C across 64 chained WMMAs.
// One fully-active wave32 (EXEC all-1s as WMMA requires). Deterministic.
__global__ __launch_bounds__(32) void final_reduce_kernel(
    const float* __restrict__ ws_masked, const float* __restrict__ ws_nll,
    float* __restrict__ d_out) {
  const int lane = threadIdx.x;
  // 32-bit A 16x4 layout: lanes 0-15 hold (M=lane, K=0/1); lanes 16-31 (M=lane-16, K=2/3)
  const int idx = 4 * (lane & 15) + 2 * (lane >> 4);

  v2f bones; bones[0] = 1.0f; bones[1] = 1.0f;
  v8f cm = {}; v8f cn = {};
#pragma unroll 4
  for (int it = 0; it < NROWS / 64; ++it) {
    const int o = it * 64 + idx;
    v2f am; am[0] = ws_masked[o]; am[1] = ws_masked[o + 1];
    v2f an; an[0] = ws_nll[o];    an[1] = ws_nll[o + 1];
    cm = __builtin_amdgcn_wmma_f32_16x16x4_f32(false, am, false, bones,
                                               (short)0, cm, false, false);
    cn = __builtin_amdgcn_wmma_f32_16x16x4_f32(false, an, false, bones,
                                               (short)0, cn, false, false);
  }
  float sm = cm[0] + cm[1] + cm[2] + cm[3] + cm[4] + cm[5] + cm[6] + cm[7];
  float sn = cn[0] + cn[1] + cn[2] + cn[3] + cn[4] + cn[5] + cn[6] + cn[7];
  sm += __shfl_xor(sm, 16, 32);          // combine M=0..7 and M=8..15 halves
  sn += __shfl_xor(sn, 16, 32);
  if (lane == 0) {
    double term1 = (double)sm / ((double)NROWS * (double)NCOLS);
    double nll   = (double)sn / (double)NROWS;
    d_out[0] = (float)(0.5 * term1 + 0.5 * nll);
  }
}

extern "C" void kernel_launch(void* const* d_in, const int* in_sizes, int n_in,
                              void* d_out, int out_size, void* d_ws, size_t ws_size,
                              hipStream_t stream) {
  const float*     output = (const float*)d_in[0];
  const long long* labels = (const long long*)d_in[1];
  float* wsf       = (float*)d_ws;
  float* ws_masked = wsf;
  float* ws_nll    = wsf + NROWS;

  row_pass_kernel<<<NROWS, BLK, 0, stream>>>(output, labels, ws_masked, ws_nll);
  final_reduce_kernel<<<1, 32, 0, stream>>>(ws_masked, ws_nll, (float*)d_out);
}